// SimpleBlock3d_17712445129242
// MI455X (gfx1250) — compile-verified
//
#include <hip/hip_runtime.h>
#include <math.h>

// ---------------------------------------------------------------------------
// FNO3D block for MI455X (gfx1250, wave32).
// fp32 end-to-end; all GEMM-shaped stages lowered to V_WMMA_F32_16X16X4_F32.
// Truncated separable DFTs replace rfftn/irfftn (only 16x16x8 corner modes
// are nonzero). Twiddle matrices are built once per block in LDS; WMMA
// A-tiles are staged through LDS with coalesced b128 loads.
// ---------------------------------------------------------------------------

typedef float v2f __attribute__((ext_vector_type(2)));
typedef float v8f __attribute__((ext_vector_type(8)));

#define WMMAF32(ACC, A, Bv) \
  (ACC) = __builtin_amdgcn_wmma_f32_16x16x4_f32(false, (A), false, (Bv), (short)0, (ACC), false, false)

constexpr int   B_   = 8, C_ = 32, X_ = 64, Y_ = 64, Z_ = 40;
constexpr int   S_   = X_ * Y_ * Z_;            // 163840 spatial points / batch
constexpr float EPS_ = 1e-5f;
constexpr float RS40 = 0.15811388300841897f;    // 1/sqrt(40)
constexpr float RS64 = 0.125f;                  // 1/sqrt(64)
constexpr float PI2  = 6.283185307179586f;

// ---------------- fc0: [points,13] x [13,32] -> planar h[b,c,s] -------------
__global__ __launch_bounds__(256) void k_fc0(const float* __restrict__ x,
                                             const float* __restrict__ w,
                                             const float* __restrict__ bias,
                                             float* __restrict__ h) {
  int wave = blockIdx.x * (blockDim.x >> 5) + (threadIdx.x >> 5);
  int lane = threadIdx.x & 31, ln = lane & 15, hi = lane >> 4;
  long p0 = (long)wave * 16;
  int b = (int)(p0 / S_);
  int s0 = (int)(p0 - (long)b * S_);
  v8f acc0 = {}, acc1 = {};
#pragma unroll
  for (int ks = 0; ks < 4; ++ks) {
    int k0 = ks * 4 + hi * 2;
    int ka = (k0 < 13) ? k0 : 12, kb = (k0 + 1 < 13) ? k0 + 1 : 12;
    float pa = (k0 < 13) ? 1.f : 0.f, pb = (k0 + 1 < 13) ? 1.f : 0.f;
    const float* xr = x + (p0 + ln) * 13;
    v2f a, b0, b1;
    a.x  = xr[ka] * pa;                 // zero A side only: 0*B == 0
    a.y  = xr[kb] * pb;
    b0.x = w[ka * 32 + ln];      b0.y = w[kb * 32 + ln];
    b1.x = w[ka * 32 + 16 + ln]; b1.y = w[kb * 32 + 16 + ln];
    WMMAF32(acc0, a, b0);
    WMMAF32(acc1, a, b1);
  }
  float bi0 = bias[ln], bi1 = bias[16 + ln];
#pragma unroll
  for (int r = 0; r < 8; ++r) {
    int m = r + hi * 8;
    h[(long)(b * C_ + ln)      * S_ + s0 + m] = acc0[r] + bi0;
    h[(long)(b * C_ + 16 + ln) * S_ + s0 + m] = acc1[r] + bi1;
  }
}

// ------------- forward Z DFT: [rows,40] x twiddle[40,16] -> zf[rows,16] -----
__global__ __launch_bounds__(256) void k_zfwd(const float* __restrict__ h,
                                              float* __restrict__ zf) {
  __shared__ float tw[16 * 40];      // [n][k] layout -> b64 B-frag loads
  __shared__ float At[8][16 * 40];   // per-wave staged A tile (20 KB)
  int tid = threadIdx.x;
  for (int i = tid; i < 640; i += 256) {
    int n = i / 40, k = i - n * 40;
    int mo = n >> 1, pim = n & 1;
    float s, c;
    __sincosf(PI2 * (float)((mo * k) % 40) / 40.f, &s, &c);
    tw[i] = RS40 * (pim ? -s : c);   // e^{-i th}: re=cos, im=-sin
  }
  int wv = tid >> 5, lane = tid & 31, ln = lane & 15, hi = lane >> 4;
  long r0 = ((long)blockIdx.x * 8 + wv) * 16;
  {                                   // coalesced b128 staging of 16x40 tile
    const float4* src = (const float4*)(h + r0 * 40);
    float4* dst = (float4*)&At[wv][0];
    for (int i = lane; i < 160; i += 32) dst[i] = src[i];
  }
  __syncthreads();
  v8f acc = {};
#pragma unroll
  for (int ks = 0; ks < 10; ++ks) {
    int k0 = ks * 4 + hi * 2;
    v2f a  = *(const v2f*)&At[wv][ln * 40 + k0];
    v2f bb = *(const v2f*)&tw[ln * 40 + k0];
    WMMAF32(acc, a, bb);
  }
#pragma unroll
  for (int r = 0; r < 8; ++r) {
    int m = r + hi * 8;
    zf[(r0 + m) * 16 + ln] = acc[r];
  }
}

// ------------- forward Y DFT (VALU): zf -> yf[bc,x,ym16,zm8,(re,im)] --------
__global__ __launch_bounds__(256) void k_yfwd(const float* __restrict__ zf,
                                              float* __restrict__ yf) {
  __shared__ float2 cs[64];
  if (threadIdx.x < 64) {
    float s, c; __sincosf(PI2 * (float)threadIdx.x / 64.f, &s, &c);
    cs[threadIdx.x] = make_float2(c, s);
  }
  __syncthreads();
  int t = blockIdx.x * blockDim.x + threadIdx.x;     // 2,097,152
  int zm = t & 7, ym = (t >> 3) & 15, xx = (t >> 7) & 63, bc = t >> 13;
  int mode = (ym < 8) ? ym : 48 + ym;                // modes 56..63
  float ar = 0.f, ai = 0.f;
  const float2* src = (const float2*)(zf + ((long)bc * 4096 + (long)xx * 64) * 16) + zm;
  for (int y = 0; y < 64; ++y) {
    float2 tc = cs[(mode * y) & 63];
    float2 v = src[(long)y * 8];
    ar += tc.x * v.x + tc.y * v.y;                   // e^{-i th}(vr+i vi)
    ai += tc.x * v.y - tc.y * v.x;
  }
  long o = (((long)bc * 64 + xx) * 16 + ym) * 8 + zm;
  ((float2*)yf)[o] = make_float2(ar * RS64, ai * RS64);
}

// ------------- forward X DFT (VALU): yf -> xf[bc,xm16,ym16,zm8,(re,im)] -----
__global__ __launch_bounds__(256) void k_xfwd(const float* __restrict__ yf,
                                              float* __restrict__ xf) {
  __shared__ float2 cs[64];
  if (threadIdx.x < 64) {
    float s, c; __sincosf(PI2 * (float)threadIdx.x / 64.f, &s, &c);
    cs[threadIdx.x] = make_float2(c, s);
  }
  __syncthreads();
  int t = blockIdx.x * blockDim.x + threadIdx.x;     // 1,048,576
  int zm = t & 7, ym = (t >> 3) & 15, xm = (t >> 7) & 15, bc = t >> 11;
  int mode = (xm < 8) ? xm : 48 + xm;
  float ar = 0.f, ai = 0.f;
  for (int xx = 0; xx < 64; ++xx) {
    float2 tc = cs[(mode * xx) & 63];
    float2 v = ((const float2*)yf)[(((long)bc * 64 + xx) * 16 + ym) * 8 + zm];
    ar += tc.x * v.x + tc.y * v.y;
    ai += tc.x * v.y - tc.y * v.x;
  }
  long o = (((long)bc * 16 + xm) * 16 + ym) * 8 + zm;
  ((float2*)xf)[o] = make_float2(ar * RS64, ai * RS64);
}

// ------------- per-mode complex 32x32 channel mix ---------------------------
__global__ __launch_bounds__(32) void k_mix(const float* __restrict__ xf,
                                            const float* __restrict__ sw,
                                            float* __restrict__ cf, int layer) {
  __shared__ float2 av[32];
  int o = threadIdx.x;
  int mm = blockIdx.x & 2047, b = blockIdx.x >> 11;
  int zm = mm & 7, ym = (mm >> 3) & 15, xm = mm >> 7;
  long ein = (((long)(b * 32 + o) * 16 + xm) * 16 + ym) * 8 + zm;
  av[o] = ((const float2*)xf)[ein];
  __syncthreads();
  int cr = (xm >= 8 ? 1 : 0) + (ym >= 8 ? 2 : 0);    // corner per reference
  int m1 = xm & 7, m2 = ym & 7;
  float outr = 0.f, outi = 0.f;
  for (int i = 0; i < 32; ++i) {
    long wi = (((((long)(layer * 4 + cr) * 32 + i) * 32 + o) * 8 + m1) * 8 + m2) * 8 + zm;
    float2 wv = ((const float2*)sw)[wi];
    outr += av[i].x * wv.x - av[i].y * wv.y;
    outi += av[i].x * wv.y + av[i].y * wv.x;
  }
  ((float2*)cf)[ein] = make_float2(outr, outi);
}

// ------------- inverse X DFT (VALU): cf -> gi[bc,x,ym,zm,(re,im)] -----------
__global__ __launch_bounds__(256) void k_ixinv(const float* __restrict__ cf,
                                               float* __restrict__ gi) {
  __shared__ float2 cs[64];
  if (threadIdx.x < 64) {
    float s, c; __sincosf(PI2 * (float)threadIdx.x / 64.f, &s, &c);
    cs[threadIdx.x] = make_float2(c, s);
  }
  __syncthreads();
  int t = blockIdx.x * blockDim.x + threadIdx.x;     // 2,097,152
  int zm = t & 7, ym = (t >> 3) & 15, xx = (t >> 7) & 63, bc = t >> 13;
  float ar = 0.f, ai = 0.f;
#pragma unroll
  for (int xm = 0; xm < 16; ++xm) {
    int mode = (xm < 8) ? xm : 48 + xm;
    float2 tc = cs[(mode * xx) & 63];
    float2 v = ((const float2*)cf)[(((long)bc * 16 + xm) * 16 + ym) * 8 + zm];
    ar += tc.x * v.x - tc.y * v.y;                   // e^{+i th}(vr+i vi)
    ai += tc.x * v.y + tc.y * v.x;
  }
  long o = (((long)bc * 64 + xx) * 16 + ym) * 8 + zm;
  ((float2*)gi)[o] = make_float2(ar * RS64, ai * RS64);
}

// ------------- inverse Y DFT (VALU): gi -> gy[rows,16] ----------------------
__global__ __launch_bounds__(256) void k_iyinv(const float* __restrict__ gi,
                                               float* __restrict__ gy) {
  __shared__ float2 cs[64];
  if (threadIdx.x < 64) {
    float s, c; __sincosf(PI2 * (float)threadIdx.x / 64.f, &s, &c);
    cs[threadIdx.x] = make_float2(c, s);
  }
  __syncthreads();
  int t = blockIdx.x * blockDim.x + threadIdx.x;     // 16,777,216
  int zm = t & 7, yy = (t >> 3) & 63, xx = (t >> 9) & 63, bc = t >> 15;
  float ar = 0.f, ai = 0.f;
#pragma unroll
  for (int ym = 0; ym < 16; ++ym) {
    int mode = (ym < 8) ? ym : 48 + ym;
    float2 tc = cs[(mode * yy) & 63];
    float2 v = ((const float2*)gi)[(((long)bc * 64 + xx) * 16 + ym) * 8 + zm];
    ar += tc.x * v.x - tc.y * v.y;
    ai += tc.x * v.y + tc.y * v.x;
  }
  long r = (long)bc * 4096 + xx * 64 + yy;
  ((float2*)gy)[r * 8 + zm] = make_float2(ar * RS64, ai * RS64);
}

// ------------- inverse Z (real): [rows,16] x twiddle[16,48] -> xa[rows,40] --
static __device__ __forceinline__ float izw(int k, int z) {
  if (z >= 40) return 0.f;
  int m = k >> 1, p = k & 1;
  if (m == 0) return p ? 0.f : RS40;                 // Re(F[0]) only
  float s, c; __sincosf(PI2 * (float)((m * z) % 40) / 40.f, &s, &c);
  return p ? (-2.f * RS40 * s) : (2.f * RS40 * c);   // 2*Re(F e^{+i th})
}
__global__ __launch_bounds__(256) void k_izinv(const float* __restrict__ gy,
                                               float* __restrict__ xa) {
  __shared__ float twz[48 * 16];     // [z][k] layout -> b64 B-frag loads
  __shared__ float Az[8][16 * 16];   // per-wave staged A tile (8 KB)
  int tid = threadIdx.x;
  for (int i = tid; i < 768; i += 256) {
    int z = i >> 4, k = i & 15;
    twz[i] = izw(k, z);
  }
  int wv = tid >> 5, lane = tid & 31, ln = lane & 15, hi = lane >> 4;
  long r0 = ((long)blockIdx.x * 8 + wv) * 16;
  {                                   // coalesced b128 staging of 16x16 tile
    const float4* src = (const float4*)(gy + r0 * 16);
    float4* dst = (float4*)&Az[wv][0];
    for (int i = lane; i < 64; i += 32) dst[i] = src[i];
  }
  __syncthreads();
#pragma unroll
  for (int nt = 0; nt < 3; ++nt) {                   // N=40 padded to 48
    int z = nt * 16 + ln;
    v8f acc = {};
#pragma unroll
    for (int ks = 0; ks < 4; ++ks) {
      int k0 = ks * 4 + hi * 2;
      v2f a  = *(const v2f*)&Az[wv][ln * 16 + k0];
      v2f bb = *(const v2f*)&twz[z * 16 + k0];
      WMMAF32(acc, a, bb);
    }
    if (z < 40) {
#pragma unroll
      for (int r = 0; r < 8; ++r) {
        int m = r + hi * 8;
        xa[(r0 + m) * 40 + z] = acc[r];
      }
    }
  }
}

// ------------- pointwise conv: w_w[32,32] x h[b,32,s] += into xa ------------
__global__ __launch_bounds__(256) void k_pconv(const float* __restrict__ h,
                                               const float* __restrict__ ww,
                                               const float* __restrict__ wb,
                                               float* __restrict__ xa, int layer) {
  int wave = blockIdx.x * (blockDim.x >> 5) + (threadIdx.x >> 5);
  int lane = threadIdx.x & 31, ln = lane & 15, hi = lane >> 4;
  long p0 = (long)wave * 16;
  int b = (int)(p0 / S_);
  int s0 = (int)(p0 - (long)b * S_);
  const float* W = ww + layer * 1024;                // [o][c]
  v8f acc0 = {}, acc1 = {};
#pragma unroll
  for (int ks = 0; ks < 8; ++ks) {
    int k0 = ks * 4 + hi * 2;
    v2f a0 = *(const v2f*)&W[ln * 32 + k0];
    v2f a1 = *(const v2f*)&W[(16 + ln) * 32 + k0];
    v2f bb;
    bb.x = h[(long)(b * 32 + k0)     * S_ + s0 + ln];  // coalesced across lanes
    bb.y = h[(long)(b * 32 + k0 + 1) * S_ + s0 + ln];
    WMMAF32(acc0, a0, bb);
    WMMAF32(acc1, a1, bb);
  }
#pragma unroll
  for (int r = 0; r < 8; ++r) {
    int m = r + hi * 8;
    long i0 = (long)(b * 32 + m)      * S_ + s0 + ln;
    long i1 = (long)(b * 32 + 16 + m) * S_ + s0 + ln;
    xa[i0] += acc0[r] + wb[layer * 32 + m];
    xa[i1] += acc1[r] + wb[layer * 32 + 16 + m];
  }
}

// ------------- BN stats: deterministic two-stage tree reduction -------------
__global__ __launch_bounds__(256) void k_partial(const float* __restrict__ xa,
                                                 float* __restrict__ part) {
  __shared__ float ssum[256], ssq[256];
  int bid = blockIdx.x, plane = bid / 40, blk = bid % 40;  // plane = b*32+c
  int b = plane >> 5, c = plane & 31;
  long base = (long)plane * S_ + (long)blk * 4096;
  float s = 0.f, q = 0.f;
#pragma unroll
  for (int i = 0; i < 16; ++i) {
    float v = xa[base + threadIdx.x + i * 256];
    s += v; q += v * v;
  }
  ssum[threadIdx.x] = s; ssq[threadIdx.x] = q;
  __syncthreads();
  for (int off = 128; off > 0; off >>= 1) {
    if ((int)threadIdx.x < off) {
      ssum[threadIdx.x] += ssum[threadIdx.x + off];
      ssq[threadIdx.x]  += ssq[threadIdx.x + off];
    }
    __syncthreads();
  }
  if (threadIdx.x == 0) {
    long pi = ((long)c * (B_ * 40) + b * 40 + blk) * 2;
    part[pi] = ssum[0]; part[pi + 1] = ssq[0];
  }
}
__global__ __launch_bounds__(256) void k_finstat(const float* __restrict__ part,
                                                 float* __restrict__ stats) {
  __shared__ float ssum[256], ssq[256];
  int c = blockIdx.x;
  float s = 0.f, q = 0.f;
  for (int i = threadIdx.x; i < B_ * 40; i += 256) {
    long pi = ((long)c * (B_ * 40) + i) * 2;
    s += part[pi]; q += part[pi + 1];
  }
  ssum[threadIdx.x] = s; ssq[threadIdx.x] = q;
  __syncthreads();
  for (int off = 128; off > 0; off >>= 1) {
    if ((int)threadIdx.x < off) {
      ssum[threadIdx.x] += ssum[threadIdx.x + off];
      ssq[threadIdx.x]  += ssq[threadIdx.x + off];
    }
    __syncthreads();
  }
  if (threadIdx.x == 0) {
    float N = (float)((long)B_ * S_);
    float mean = ssum[0] / N;
    float var = ssq[0] / N - mean * mean;
    stats[c * 2] = mean;
    stats[c * 2 + 1] = rsqrtf(fmaxf(var, 0.f) + EPS_);
  }
}

// ------------- BN apply (+ReLU for layers 0..2), float4 path ----------------
__global__ __launch_bounds__(256) void k_bn(const float* __restrict__ xa,
                                            const float* __restrict__ stats,
                                            const float* __restrict__ g,
                                            const float* __restrict__ bb,
                                            float* __restrict__ h, int layer,
                                            int dorelu) {
  long e4 = (long)(blockIdx.x * blockDim.x + threadIdx.x);
  int c = (int)((e4 * 4 / S_) & 31);                 // S_ % 4 == 0
  float mean = stats[c * 2], rstd = stats[c * 2 + 1];
  float gg = g[layer * 32 + c] * rstd, bv = bb[layer * 32 + c];
  float4 v = ((const float4*)xa)[e4];
  v.x = (v.x - mean) * gg + bv;
  v.y = (v.y - mean) * gg + bv;
  v.z = (v.z - mean) * gg + bv;
  v.w = (v.w - mean) * gg + bv;
  if (dorelu) {
    v.x = fmaxf(v.x, 0.f); v.y = fmaxf(v.y, 0.f);
    v.z = fmaxf(v.z, 0.f); v.w = fmaxf(v.w, 0.f);
  }
  ((float4*)h)[e4] = v;
}

// ------------- fused fc1(32->128)+ReLU+fc2(128->1) --------------------------
__global__ __launch_bounds__(256) void k_fc12(const float* __restrict__ h,
                                              const float* __restrict__ w1,
                                              const float* __restrict__ b1,
                                              const float* __restrict__ w2,
                                              const float* __restrict__ b2,
                                              float* __restrict__ out) {
  int wave = blockIdx.x * (blockDim.x >> 5) + (threadIdx.x >> 5);
  int lane = threadIdx.x & 31, ln = lane & 15, hi = lane >> 4;
  long p0 = (long)wave * 16;
  int b = (int)(p0 / S_);
  int s0 = (int)(p0 - (long)b * S_);
  v2f aa[8];
#pragma unroll
  for (int ks = 0; ks < 8; ++ks) {
    int k0 = ks * 4 + hi * 2;
    aa[ks].x = h[(long)(b * 32 + k0)     * S_ + s0 + ln];
    aa[ks].y = h[(long)(b * 32 + k0 + 1) * S_ + s0 + ln];
  }
  float part[8];
#pragma unroll
  for (int r = 0; r < 8; ++r) part[r] = 0.f;
#pragma unroll
  for (int nt = 0; nt < 8; ++nt) {                   // N=128
    int n = nt * 16 + ln;
    v8f acc = {};
#pragma unroll
    for (int ks = 0; ks < 8; ++ks) {
      int k0 = ks * 4 + hi * 2;
      v2f bb;
      bb.x = w1[k0 * 128 + n];
      bb.y = w1[(k0 + 1) * 128 + n];
      WMMAF32(acc, aa[ks], bb);
    }
    float bv = b1[n], w2v = w2[n];
#pragma unroll
    for (int r = 0; r < 8; ++r) {
      float v = fmaxf(acc[r] + bv, 0.f);             // ReLU(fc1)
      part[r] += v * w2v;                            // fc2 dot
    }
  }
#pragma unroll
  for (int r = 0; r < 8; ++r) {
    float p = part[r];
    for (int off = 1; off < 16; off <<= 1) p += __shfl_xor(p, off, 32);
    if (ln == 0) {
      int m = r + hi * 8;
      out[(long)b * S_ + s0 + m] = p + b2[0];
    }
  }
}

// ---------------------------------------------------------------------------
extern "C" void kernel_launch(void* const* d_in, const int* in_sizes, int n_in,
                              void* d_out, int out_size, void* d_ws, size_t ws_size,
                              hipStream_t stream) {
  const float* x     = (const float*)d_in[0];
  const float* fc0_w = (const float*)d_in[1];
  const float* fc0_b = (const float*)d_in[2];
  const float* spec  = (const float*)d_in[3];
  const float* w_w   = (const float*)d_in[4];
  const float* w_b   = (const float*)d_in[5];
  const float* bn_g  = (const float*)d_in[6];
  const float* bn_b  = (const float*)d_in[7];
  const float* fc1_w = (const float*)d_in[8];
  const float* fc1_b = (const float*)d_in[9];
  const float* fc2_w = (const float*)d_in[10];
  const float* fc2_b = (const float*)d_in[11];
  (void)in_sizes; (void)n_in; (void)out_size; (void)ws_size;

  float* ws = (float*)d_ws;
  float* h     = ws;                      // 41,943,040  (168 MB) planar [b,c,s]
  float* xa    = ws + 41943040L;          // 41,943,040  x1+x2 accumulator
  float* zf    = ws + 83886080L;          // 33,554,432  Z spectrum / gy (reused)
  float* yf    = ws + 117440512L;         //  8,388,608  Y spectrum / gi (reused)
  float* xf    = ws + 125829120L;         //  2,097,152  corner modes
  float* cf    = ws + 127926272L;         //  2,097,152  mixed modes
  float* part  = ws + 130023424L;         //     20,480  BN partials
  float* stats = ws + 130043904L;         //         64  BN mean / rstd
  // total 130,043,968 floats ~= 496 MiB of d_ws

  dim3 blk(256);
  k_fc0<<<10240, blk, 0, stream>>>(x, fc0_w, fc0_b, h);
  for (int l = 0; l < 4; ++l) {
    k_zfwd <<<8192,  blk, 0, stream>>>(h, zf);
    k_yfwd <<<8192,  blk, 0, stream>>>(zf, yf);
    k_xfwd <<<4096,  blk, 0, stream>>>(yf, xf);
    k_mix  <<<16384, dim3(32), 0, stream>>>(xf, spec, cf, l);
    k_ixinv<<<8192,  blk, 0, stream>>>(cf, yf);      // yf reused as gi
    k_iyinv<<<65536, blk, 0, stream>>>(yf, zf);      // zf reused as gy
    k_izinv<<<8192,  blk, 0, stream>>>(zf, xa);      // x1
    k_pconv<<<10240, blk, 0, stream>>>(h, w_w, w_b, xa, l);  // += x2 + bias
    k_partial<<<10240, blk, 0, stream>>>(xa, part);
    k_finstat<<<32,    blk, 0, stream>>>(part, stats);
    k_bn   <<<40960, blk, 0, stream>>>(xa, stats, bn_g, bn_b, h, l, (l < 3) ? 1 : 0);
  }
  k_fc12<<<10240, blk, 0, stream>>>(h, fc1_w, fc1_b, fc2_w, fc2_b, (float*)d_out);
}